// DiMPL2SteepestDescentGN_25177098289205
// MI455X (gfx1250) — compile-verified
//
#include <hip/hip_runtime.h>
#include <hip/hip_bf16.h>

// ---------------------------------------------------------------------------
// DiMP L2 steepest-descent GN optimizer for MI455X (gfx1250, wave32, WMMA)
//
// Shapes: N=3, S=64, C=512, H=W=22, fH=fW=4, out=23x23, NUM_ITER=5
// Heavy ops recast as WMMA f32 16x16x4 GEMMs (16 filter taps = N dimension).
// Filters / residual maps kept in pair-interleaved layouts so B fragments are
// single coalesced b64 loads; A fragments are branch-free (index clamping).
// ---------------------------------------------------------------------------

typedef float v2f __attribute__((ext_vector_type(2)));
typedef float v8f __attribute__((ext_vector_type(8)));

#define NB    3
#define SB    64
#define CB    512
#define HB    22
#define WB    22
#define NPIX  (HB * WB)        // 484
#define NP2   (NPIX / 2)       // 242
#define OUT0  23
#define OUT1  23
#define NSC   (OUT0 * OUT1)    // 529
#define TAPS  16
#define WELEM (SB * CB * TAPS) // 524288 floats per filter tensor
#define NTILES 31              // ceil(484/16)

__device__ __forceinline__ v8f wmma4(v2f a, v2f b, v8f c)
{
    return __builtin_amdgcn_wmma_f32_16x16x4_f32(
        false, a, false, b, (short)0, c, false, false);
}

// pair-interleaved filter layout: float index = s*8192 + (c>>1)*32 + tap*2 + (c&1)
__device__ __forceinline__ size_t pair_idx(int s, int c, int tap)
{
    return (size_t)s * (CB * TAPS) + (size_t)(c >> 1) * 32 + tap * 2 + (c & 1);
}

// ---------------------------------------------------------------------------
// GEMM1/GEMM3: P[n][s][pix][tap] = sum_c feat[n][s][c][pix] * w[s][c][tap]
// One wave per (n, s, 16-pixel tile). M=pixels, N=taps, K=C (128 K4 steps).
// wpair: filter in pair layout. Branch-free (pixel clamp), double buffered.
// ---------------------------------------------------------------------------
__global__ __launch_bounds__(32) void gemm_feat_filter(
    const float* __restrict__ feat, const float* __restrict__ wpair,
    float* __restrict__ P)
{
    int bid  = blockIdx.x;
    int tile = bid % NTILES;
    int ns   = bid / NTILES;        // n*SB + s
    int s    = ns % SB;
    int lane = threadIdx.x;
    int m    = lane & 15;           // M (pixel) / N (tap) sub-index
    int kh   = lane >> 4;           // K half: lanes 16-31 hold K+2

    int pix = tile * 16 + m;
    if (pix >= NPIX) pix = NPIX - 1;  // clamp: garbage rows masked at store

    const float* fb = feat + ((size_t)ns * CB) * NPIX + pix;
    const v2f*   wb = (const v2f*)(wpair + (size_t)s * CB * TAPS) + m;

    v8f acc = {};
    v2f a, b;
    {
        int ka = 2 * kh;
        a.x = fb[(size_t)ka * NPIX];
        a.y = fb[(size_t)(ka + 1) * NPIX];
        b   = wb[(size_t)kh * TAPS];
    }
#pragma unroll 2
    for (int c0 = 4; c0 < CB; c0 += 4) {
        int kn = c0 + 2 * kh;
        v2f an, bn;
        an.x = fb[(size_t)kn * NPIX];
        an.y = fb[(size_t)(kn + 1) * NPIX];
        bn   = wb[(size_t)(c0 / 2 + kh) * TAPS];
        acc  = wmma4(a, b, acc);
        a = an; b = bn;
    }
    acc = wmma4(a, b, acc);

    // D layout: VGPR r -> M = r + 8*kh, N = m
    float* pb = P + ((size_t)ns * NPIX) * TAPS;
#pragma unroll
    for (int r = 0; r < 8; ++r) {
        int p = tile * 16 + r + 8 * kh;
        if (p < NPIX) pb[(size_t)p * TAPS + m] = acc[r];
    }
}

// ---------------------------------------------------------------------------
// GEMM2: wgrad[s][c][tap] = sum_n sum_pix feat[n][s][c][pix]*R[n][s][pix][tap]
//        + reg * w[s][c][tap]            (outputs in pair layout)
// Also accumulates a_num[s] += |wgrad|^2 and lossSlot += reg * |w|^2.
// One wave per (s, 16-channel tile). M=channels, N=taps, K=N*NPIX=1452.
// ---------------------------------------------------------------------------
__global__ __launch_bounds__(32) void gemm_wgrad(
    const float* __restrict__ feat, const float* __restrict__ RtP,
    const float* __restrict__ wbuf, const float* __restrict__ freg,
    float* __restrict__ wgrad, float* __restrict__ a_num,
    float* __restrict__ lossSlot)
{
    int bid  = blockIdx.x;
    int ct   = bid % (CB / 16);
    int s    = bid / (CB / 16);
    int lane = threadIdx.x;
    int m    = lane & 15;
    int kh   = lane >> 4;
    int c_a  = ct * 16 + m;

    v8f acc = {};
    for (int n = 0; n < NB; ++n) {
        // A: feat row for channel c_a; kp even -> aligned b64 pair loads
        const v2f* fb = (const v2f*)(feat +
                        (((size_t)n * SB + s) * CB + c_a) * NPIX);
        // B: shifted residual map in pair layout [p2][tap][2]
        const v2f* rb = (const v2f*)(RtP +
                        (((size_t)n * SB + s) * NPIX) * TAPS) + m;

        v2f a = fb[kh];                       // pix pair (2kh, 2kh+1)
        v2f b = rb[(size_t)kh * TAPS];
#pragma unroll 2
        for (int p0 = 4; p0 < NPIX; p0 += 4) {   // 484 % 4 == 0
            int k2 = p0 / 2 + kh;
            v2f an = fb[k2];
            v2f bn = rb[(size_t)k2 * TAPS];
            acc = wmma4(a, b, acc);
            a = an; b = bn;
        }
        acc = wmma4(a, b, acc);
    }

    float reg  = fmaxf(freg[0] * freg[0], 1e-6f);
    float gsum = 0.0f, wsum = 0.0f;
    // lane holds c = ct*16 + 8*kh + r (r=0..7), tap = m -> consecutive c:
    // emit pair-layout b64 stores.
    v2f*       wgp = (v2f*)(wgrad + (size_t)s * CB * TAPS);
    const v2f* wvp = (const v2f*)(wbuf + (size_t)s * CB * TAPS);
#pragma unroll
    for (int r = 0; r < 8; r += 2) {
        int c2     = ct * 8 + 4 * kh + r / 2;   // (c>>1) for the even c
        size_t o   = (size_t)c2 * TAPS + m;
        v2f wcur   = wvp[o];
        v2f g;
        g.x = acc[r]     + reg * wcur.x;
        g.y = acc[r + 1] + reg * wcur.y;
        wgp[o] = g;
        gsum += g.x * g.x + g.y * g.y;
        wsum += wcur.x * wcur.x + wcur.y * wcur.y;
    }
    atomicAdd(a_num + s, gsum);
    atomicAdd(lossSlot, reg * wsum);
}

// ---------------------------------------------------------------------------
// Gather P -> scores, build label/masks from bb, residual, rm, smask, loss.
// ---------------------------------------------------------------------------
__device__ __forceinline__ float gather_score(const float* __restrict__ pb,
                                              int y, int x)
{
    float sc = 0.0f;
#pragma unroll
    for (int ky = 0; ky < 4; ++ky) {
        int py = y + ky - 2;
        if (py < 0 || py >= HB) continue;
#pragma unroll
        for (int kx = 0; kx < 4; ++kx) {
            int px = x + kx - 2;
            if (px < 0 || px >= WB) continue;
            sc += pb[(size_t)(py * WB + px) * TAPS + (ky * 4 + kx)];
        }
    }
    return sc;
}

__global__ void residual_kernel(
    const float* __restrict__ P, const float* __restrict__ bb,
    const float* __restrict__ sweight, float* __restrict__ rm,
    float* __restrict__ smask, float* __restrict__ lossSlot)
{
    int idx = blockIdx.x * blockDim.x + threadIdx.x;
    if (idx >= NB * SB * NSC) return;
    int pix = idx % NSC;
    int ns  = idx / NSC;
    int s   = ns % SB;
    int n   = ns / SB;
    int y   = pix / OUT1, x = pix % OUT1;

    float sc = gather_score(P + ((size_t)ns * NPIX) * TAPS, y, x);

    const float* b4 = bb + ((size_t)n * SB + s) * 4;
    float cy = (b4[1] + 0.5f * b4[3]) * (1.0f / 16.0f);
    float cx = (b4[0] + 0.5f * b4[2]) * (1.0f / 16.0f);
    const float inv2s2 = 1.0f / (2.0f * 0.9f * 0.9f);
    float g = __expf(-inv2s2 * ((float)y - cy) * ((float)y - cy)) *
              __expf(-inv2s2 * ((float)x - cx) * ((float)x - cx));
    float tm    = (g > 0.05f) ? 1.0f : 0.0f;
    float label = g * tm;

    float sa = (tm > 0.0f) ? sc : fmaxf(sc, 0.0f);           // hinged activation
    float sm = (tm > 0.0f) ? 1.0f : ((sc > 0.0f) ? 1.0f : 0.0f);
    float sw = sqrtf(sweight[(size_t)n * SB + s]);
    float resid = sw * (sa - label);

    atomicAdd(lossSlot, resid * resid);
    rm[idx]    = sm * sw * resid;   // sample weight applied twice, as reference
    smask[idx] = sm;
}

// RtP[n][s][p2][tap][2] = rm shifted by tap (0 if OOB), pair-interleaved
__global__ void build_rt_kernel(const float* __restrict__ rm,
                                float* __restrict__ RtP)
{
    int idx = blockIdx.x * blockDim.x + threadIdx.x;
    if (idx >= NB * SB * NPIX) return;
    int pix = idx % NPIX;
    int ns  = idx / NPIX;
    int py  = pix / WB, px = pix % WB;
    const float* r = rm + (size_t)ns * NSC;
    float* rt = RtP + (((size_t)ns * NPIX + (pix >> 1) * 2) * TAPS) + (pix & 1);
#pragma unroll
    for (int ky = 0; ky < 4; ++ky) {
        int y = py - ky + 2;
#pragma unroll
        for (int kx = 0; kx < 4; ++kx) {
            int x = px - kx + 2;
            rt[(ky * 4 + kx) * 2] =
                (y >= 0 && y < OUT0 && x >= 0 && x < OUT1) ? r[y * OUT1 + x]
                                                           : 0.0f;
        }
    }
}

// sgrad reduction: a_den[s] += (sw * smask * apply_filter(feat, wgrad))^2
__global__ void sgrad_kernel(
    const float* __restrict__ P2, const float* __restrict__ smask,
    const float* __restrict__ sweight, float* __restrict__ a_den)
{
    int idx = blockIdx.x * blockDim.x + threadIdx.x;
    if (idx >= NB * SB * NSC) return;
    int pix = idx % NSC;
    int ns  = idx / NSC;
    int s   = ns % SB;
    int n   = ns / SB;
    int y   = pix / OUT1, x = pix % OUT1;

    float sc = gather_score(P2 + ((size_t)ns * NPIX) * TAPS, y, x);
    float sw = sqrtf(sweight[(size_t)n * SB + s]);
    float sg = sw * smask[idx] * sc;
    atomicAdd(a_den + s, sg * sg);
}

// w -= step * alpha[s] * wgrad ; emit iterate (and final w on last iteration)
__global__ void update_kernel(
    float* __restrict__ wbuf, const float* __restrict__ wgrad,
    const float* __restrict__ a_num, const float* __restrict__ a_den,
    const float* __restrict__ logstep, const float* __restrict__ freg,
    float* __restrict__ out_iter, float* __restrict__ out_w, int write_final)
{
    int i = blockIdx.x * blockDim.x + threadIdx.x;
    if (i >= WELEM) return;
    int s   = i / (CB * TAPS);
    int rr  = i % (CB * TAPS);
    int c   = rr / TAPS;
    int tap = rr % TAPS;
    size_t pidx = pair_idx(s, c, tap);

    float reg   = fmaxf(freg[0] * freg[0], 1e-6f);
    float step  = __expf(logstep[0]);
    float num   = a_num[s];
    float den   = fmaxf(a_den[s] + reg * num, 1e-8f);
    float alpha = num / den;
    float nw = wbuf[pidx] - step * alpha * wgrad[pidx];
    wbuf[pidx]  = nw;
    out_iter[i] = nw;
    if (write_final) out_w[i] = nw;
}

__global__ void init_kernel(const float* __restrict__ weights,
                            float* __restrict__ wbuf,
                            float* __restrict__ out_iter0,
                            float* __restrict__ lossAcc)
{
    int i = blockIdx.x * blockDim.x + threadIdx.x;
    if (i < WELEM) {
        int s   = i / (CB * TAPS);
        int rr  = i % (CB * TAPS);
        int c   = rr / TAPS;
        int tap = rr % TAPS;
        float v = weights[i];
        wbuf[pair_idx(s, c, tap)] = v;
        out_iter0[i] = v;
    }
    if (blockIdx.x == 0 && threadIdx.x < 8) lossAcc[threadIdx.x] = 0.0f;
}

__global__ void zero_sacc_kernel(float* __restrict__ a_num,
                                 float* __restrict__ a_den)
{
    int i = threadIdx.x;
    if (i < SB) { a_num[i] = 0.0f; a_den[i] = 0.0f; }
}

__global__ void finalize_kernel(const float* __restrict__ lossAcc,
                                float* __restrict__ out_loss)
{
    int i = threadIdx.x;
    if (i < 6) out_loss[i] = lossAcc[i] * (1.0f / (float)SB);
}

// ---------------------------------------------------------------------------
extern "C" void kernel_launch(void* const* d_in, const int* in_sizes, int n_in,
                              void* d_out, int out_size, void* d_ws, size_t ws_size,
                              hipStream_t stream)
{
    const float* weights = (const float*)d_in[0];   // (S,C,4,4)
    const float* feat    = (const float*)d_in[1];   // (N,S,C,H,W)
    const float* bb      = (const float*)d_in[2];   // (N,S,4)
    const float* sweight = (const float*)d_in[3];   // (N,S)
    const float* logstep = (const float*)d_in[4];   // (1,)
    const float* freg    = (const float*)d_in[5];   // (1,)

    float* out      = (float*)d_out;
    float* out_w    = out;                          // WELEM
    float* out_iter = out + WELEM;                  // 6*WELEM
    float* out_loss = out + (size_t)WELEM * 7;      // 6

    float* ws = (float*)d_ws;
    size_t off = 0;
    float* wbuf    = ws + off; off += WELEM;                         // pair layout
    float* wgrad   = ws + off; off += WELEM;                         // pair layout
    float* P       = ws + off; off += (size_t)NB * SB * NPIX * TAPS;
    float* RtP     = ws + off; off += (size_t)NB * SB * NPIX * TAPS; // pair layout
    float* rm      = ws + off; off += (size_t)NB * SB * NSC;
    float* smask   = ws + off; off += (size_t)NB * SB * NSC;
    float* lossAcc = ws + off; off += 8;
    float* a_num   = ws + off; off += 64;
    float* a_den   = ws + off; off += 64;
    (void)ws_size; (void)in_sizes; (void)n_in; (void)out_size;

    const int gemm1_blocks = NB * SB * NTILES;          // 5952 waves
    const int gemm2_blocks = SB * (CB / 16);            // 2048 waves
    const int score_blocks = (NB * SB * NSC + 255) / 256;
    const int rt_blocks    = (NB * SB * NPIX + 255) / 256;
    const int w_blocks     = (WELEM + 255) / 256;

    init_kernel<<<w_blocks, 256, 0, stream>>>(weights, wbuf, out_iter, lossAcc);

    for (int it = 0; it < 5; ++it) {
        zero_sacc_kernel<<<1, 128, 0, stream>>>(a_num, a_den);
        // scores = apply_filter(feat, w)
        gemm_feat_filter<<<gemm1_blocks, 32, 0, stream>>>(feat, wbuf, P);
        residual_kernel<<<score_blocks, 256, 0, stream>>>(
            P, bb, sweight, rm, smask, lossAcc + it);
        build_rt_kernel<<<rt_blocks, 256, 0, stream>>>(rm, RtP);
        // wgrad = apply_feat_transpose(feat, rm) + reg*w  (+ a_num, +reg|w|^2)
        gemm_wgrad<<<gemm2_blocks, 32, 0, stream>>>(
            feat, RtP, wbuf, freg, wgrad, a_num, lossAcc + it);
        // sgrad = sw * smask * apply_filter(feat, wgrad)  -> a_den
        gemm_feat_filter<<<gemm1_blocks, 32, 0, stream>>>(feat, wgrad, P);
        sgrad_kernel<<<score_blocks, 256, 0, stream>>>(P, smask, sweight, a_den);
        update_kernel<<<w_blocks, 256, 0, stream>>>(
            wbuf, wgrad, a_num, a_den, logstep, freg,
            out_iter + (size_t)(it + 1) * WELEM, out_w, (it == 4) ? 1 : 0);
    }

    // Final loss on updated w (residual_kernel's extra rm/smask writes are
    // scratch-only and harmless).
    gemm_feat_filter<<<gemm1_blocks, 32, 0, stream>>>(feat, wbuf, P);
    residual_kernel<<<score_blocks, 256, 0, stream>>>(
        P, bb, sweight, rm, smask, lossAcc + 5);

    finalize_kernel<<<1, 32, 0, stream>>>(lossAcc, out_loss);
}